// GNNModel_26242250179342
// MI455X (gfx1250) — compile-verified
//
#include <hip/hip_runtime.h>
#include <hip/hip_bf16.h>

#define NEG_SLOPE 0.2f

typedef __attribute__((ext_vector_type(2))) float v2f;
typedef __attribute__((ext_vector_type(8))) float v8f;

#if __has_builtin(__builtin_amdgcn_wmma_f32_16x16x4_f32)
#define HAVE_WMMA_F32X4 1
#else
#define HAVE_WMMA_F32X4 0
#endif

// ---- monotone float<->uint key for atomicMax-based segment_max ----
__device__ __forceinline__ unsigned int f2key(float f) {
    unsigned int b = __float_as_uint(f);
    return (b & 0x80000000u) ? ~b : (b | 0x80000000u);
}
__device__ __forceinline__ float key2f(unsigned int k) {
    unsigned int b = (k & 0x80000000u) ? (k ^ 0x80000000u) : ~k;
    return __uint_as_float(b);
}

// ============================================================================
// conv1 node transform: xl1 = x @ Wl1^T + bl1 ; xr1 = x @ Wr1^T + br1  (K=3)
// ============================================================================
__global__ void k_transform1(const float* __restrict__ x,
                             const float* __restrict__ Wl, const float* __restrict__ bl,
                             const float* __restrict__ Wr, const float* __restrict__ br,
                             float* __restrict__ xl, float* __restrict__ xr, int n)
{
    int idx = blockIdx.x * blockDim.x + threadIdx.x;
    if (idx >= n * 256) return;
    int node = idx >> 8, j = idx & 255;
    float x0 = x[node * 3 + 0], x1 = x[node * 3 + 1], x2 = x[node * 3 + 2];
    xl[idx] = fmaf(Wl[j * 3 + 0], x0, fmaf(Wl[j * 3 + 1], x1, fmaf(Wl[j * 3 + 2], x2, bl[j])));
    xr[idx] = fmaf(Wr[j * 3 + 0], x0, fmaf(Wr[j * 3 + 1], x1, fmaf(Wr[j * 3 + 2], x2, br[j])));
}

// ============================================================================
// conv2 node transform via V_WMMA_F32_16X16X4_F32:
//   out[m, col] = sum_k h1[m,k] * W[col,k] + bias[col]   (K=256, out dim 128)
// 4 waves / 128-thread block, one 16x16 tile per wave (wave-uniform tile id,
// so EXEC is all-1s around the WMMA). Out-of-range A rows are CLAMPED to a
// valid row: they only pollute D rows that are never stored, keeping the
// inner loop free of per-iteration selects (pure load + wmma).
// A layout: lane = M (mod 16), {v0,v1} = K {0,1} (lanes 0-15) / {2,3} (16-31)
// B layout: lane = N (mod 16), same K half-split
// D layout: VGPR r -> row r (lanes 0-15) / row r+8 (lanes 16-31), N = lane&15
// ============================================================================
__global__ __launch_bounds__(128)
void k_transform2_wmma(const float* __restrict__ h1,
                       const float* __restrict__ Wl, const float* __restrict__ bl,
                       const float* __restrict__ Wr, const float* __restrict__ br,
                       float* __restrict__ xl2, float* __restrict__ xr2,
                       int n, int mtiles)
{
    const int lane = threadIdx.x & 31;
    const int tile = blockIdx.x * 4 + (threadIdx.x >> 5);   // wave-uniform
    const int ntiles_total = mtiles * 8 * 2;
    if (tile >= ntiles_total) return;                       // whole-wave exit

    const int which = tile / (mtiles * 8);
    const int rem   = tile - which * (mtiles * 8);
    const int nt    = rem / mtiles;
    const int mt    = rem - nt * mtiles;

    const float* W    = which ? Wr  : Wl;
    const float* bias = which ? br  : bl;
    float*       out  = which ? xr2 : xl2;

    const int row   = mt * 16 + (lane & 15);       // A-matrix row (node)
    const int col   = nt * 16 + (lane & 15);       // B-matrix col (out channel)
    const int khalf = (lane >> 4) * 2;             // 0 or 2
    const int rowc  = row < n ? row : (n - 1);     // clamp: unstored rows only
    const int mbase = mt * 16 + ((lane >> 4) ? 8 : 0);

    const float2* arow = (const float2*)(h1 + (size_t)rowc * 256);
    const float2* brow = (const float2*)(W + (size_t)col * 256);

    v8f acc = {};
#if HAVE_WMMA_F32X4
    #pragma unroll 8
    for (int k = 0; k < 256; k += 4) {
        float2 af = arow[(k + khalf) >> 1];
        float2 bf = brow[(k + khalf) >> 1];
        v2f a = {af.x, af.y};
        v2f b = {bf.x, bf.y};
        acc = __builtin_amdgcn_wmma_f32_16x16x4_f32(
            false, a, false, b, (short)0, acc, false, false);
    }
#else
    // scalar fallback: each lane computes its 8 D elements directly
    for (int r = 0; r < 8; ++r) {
        int m = mbase + r;
        if (m < n) {
            float s = 0.f;
            for (int k = 0; k < 256; ++k)
                s = fmaf(h1[(size_t)m * 256 + k], W[(size_t)col * 256 + k], s);
            acc[r] = s;
        }
    }
#endif
    const float bv = bias[col];
    #pragma unroll
    for (int r = 0; r < 8; ++r) {
        int m = mbase + r;
        if (m < n) out[(size_t)m * 128 + col] = acc[r] + bv;
    }
}

// ============================================================================
// Edge pass 1: e = sum_c leakyrelu(xl[src]+xr[dst]) * att ; segment max
// ============================================================================
template <int H>
__global__ void k_edge_attn_max(const int* __restrict__ srcA, const int* __restrict__ dstA,
                                int E, int n,
                                const float* __restrict__ xl, const float* __restrict__ xr,
                                const float* __restrict__ att,
                                float* __restrict__ eout, unsigned int* __restrict__ emax)
{
    int idx = blockIdx.x * blockDim.x + threadIdx.x;
    int total = (E + n) * H;
    if (idx >= total) return;
    int e = idx / H, h = idx - e * H;
    int s, d;
    if (e < E) { s = srcA[e]; d = dstA[e]; } else { s = d = e - E; }
    const float4* xls = (const float4*)(xl + ((size_t)s * H + h) * 64);
    const float4* xrd = (const float4*)(xr + ((size_t)d * H + h) * 64);
    const float4* at  = (const float4*)(att + h * 64);
    float acc = 0.f;
    #pragma unroll
    for (int c = 0; c < 16; ++c) {
        float4 a = xls[c], b = xrd[c], w = at[c];
        float m;
        m = a.x + b.x; m = m > 0.f ? m : NEG_SLOPE * m; acc = fmaf(m, w.x, acc);
        m = a.y + b.y; m = m > 0.f ? m : NEG_SLOPE * m; acc = fmaf(m, w.y, acc);
        m = a.z + b.z; m = m > 0.f ? m : NEG_SLOPE * m; acc = fmaf(m, w.z, acc);
        m = a.w + b.w; m = m > 0.f ? m : NEG_SLOPE * m; acc = fmaf(m, w.w, acc);
    }
    eout[idx] = acc;
    atomicMax(&emax[(size_t)d * H + h], f2key(acc));
}

// ============================================================================
// Edge pass 2: ex = exp(e - max[dst]); store ex over e; segment-sum denom
// ============================================================================
template <int H>
__global__ void k_edge_exp_sum(const int* __restrict__ dstA, int E, int n,
                               float* __restrict__ e_io,
                               const unsigned int* __restrict__ emax,
                               float* __restrict__ den)
{
    int idx = blockIdx.x * blockDim.x + threadIdx.x;
    int total = (E + n) * H;
    if (idx >= total) return;
    int e = idx / H, h = idx - e * H;
    int d = (e < E) ? dstA[e] : (e - E);
    float mx = key2f(emax[(size_t)d * H + h]);
    float ex = expf(e_io[idx] - mx);
    e_io[idx] = ex;
    atomicAdd(&den[(size_t)d * H + h], ex);
}

// ============================================================================
// Edge pass 3: agg[dst] += (ex/den[dst]) * xl[src]
// ============================================================================
template <int H>
__global__ void k_edge_aggregate(const int* __restrict__ srcA, const int* __restrict__ dstA,
                                 int E, int n,
                                 const float* __restrict__ xl,
                                 const float* __restrict__ e_ex,
                                 const float* __restrict__ den,
                                 float* __restrict__ agg)
{
    int idx = blockIdx.x * blockDim.x + threadIdx.x;
    int total = (E + n) * H;
    if (idx >= total) return;
    int e = idx / H, h = idx - e * H;
    int s, d;
    if (e < E) { s = srcA[e]; d = dstA[e]; } else { s = d = e - E; }
    float alpha = e_ex[idx] / den[(size_t)d * H + h];
    const float4* xls = (const float4*)(xl + ((size_t)s * H + h) * 64);
    float* o = agg + ((size_t)d * H + h) * 64;
    #pragma unroll
    for (int c = 0; c < 16; ++c) {
        float4 v = xls[c];
        atomicAdd(&o[c * 4 + 0], alpha * v.x);
        atomicAdd(&o[c * 4 + 1], alpha * v.y);
        atomicAdd(&o[c * 4 + 2], alpha * v.z);
        atomicAdd(&o[c * 4 + 3], alpha * v.w);
    }
}

// h1 = relu(agg1 + b1), in place on agg1 ([n,256])
__global__ void k_bias_relu(float* __restrict__ io, const float* __restrict__ bias,
                            int total, int stride)
{
    int idx = blockIdx.x * blockDim.x + threadIdx.x;
    if (idx >= total) return;
    float v = io[idx] + bias[idx % stride];
    io[idx] = v > 0.f ? v : 0.f;
}

// h2 = relu(mean_heads(agg2) + b2) : agg2 [n,2,64] -> h2 [n,64]
__global__ void k_finalize2(const float* __restrict__ agg2, const float* __restrict__ b2,
                            float* __restrict__ h2, int n)
{
    int idx = blockIdx.x * blockDim.x + threadIdx.x;
    if (idx >= n * 64) return;
    int node = idx >> 6, c = idx & 63;
    float v = 0.5f * (agg2[(size_t)node * 128 + c] + agg2[(size_t)node * 128 + 64 + c]) + b2[c];
    h2[idx] = v > 0.f ? v : 0.f;
}

// roles = h2 @ Wrole^T + brole  -> d_out[0 .. n*4)
__global__ void k_roles(const float* __restrict__ h2, const float* __restrict__ Wrole,
                        const float* __restrict__ brole, float* __restrict__ out, int n)
{
    int idx = blockIdx.x * blockDim.x + threadIdx.x;
    if (idx >= n * 4) return;
    int node = idx >> 2, r = idx & 3;
    const float4* hv = (const float4*)(h2 + (size_t)node * 64);
    const float4* wv = (const float4*)(Wrole + r * 64);
    float acc = 0.f;
    #pragma unroll
    for (int c = 0; c < 16; ++c) {
        float4 a = hv[c], w = wv[c];
        acc = fmaf(a.x, w.x, acc); acc = fmaf(a.y, w.y, acc);
        acc = fmaf(a.z, w.z, acc); acc = fmaf(a.w, w.w, acc);
    }
    out[idx] = acc + brole[r];
}

// global mean pool accumulation
__global__ void k_pool(const float* __restrict__ h2, const int* __restrict__ batch,
                       float* __restrict__ gsum, float* __restrict__ gcnt, int n)
{
    int idx = blockIdx.x * blockDim.x + threadIdx.x;
    if (idx >= n * 64) return;
    int node = idx >> 6, c = idx & 63;
    int g = batch[node];
    atomicAdd(&gsum[g * 64 + c], h2[idx]);
    if (c == 0) atomicAdd(&gcnt[g], 1.0f);
}

// graph_emb = (gsum/max(cnt,1)) @ Wproj^T + bproj -> d_out[n*4 ..)
__global__ void k_proj(const float* __restrict__ gsum, const float* __restrict__ gcnt,
                       const float* __restrict__ Wproj, const float* __restrict__ bproj,
                       float* __restrict__ out, int G)
{
    int idx = blockIdx.x * blockDim.x + threadIdx.x;
    if (idx >= G * 64) return;
    int g = idx >> 6, o = idx & 63;
    float cnt = gcnt[g];
    float inv = 1.0f / (cnt > 1.0f ? cnt : 1.0f);
    const float4* sv = (const float4*)(gsum + g * 64);
    const float4* wv = (const float4*)(Wproj + o * 64);
    float acc = 0.f;
    #pragma unroll
    for (int c = 0; c < 16; ++c) {
        float4 s = sv[c], w = wv[c];
        acc = fmaf(s.x, w.x, acc); acc = fmaf(s.y, w.y, acc);
        acc = fmaf(s.z, w.z, acc); acc = fmaf(s.w, w.w, acc);
    }
    out[idx] = acc * inv + bproj[o];
}

// ============================================================================
extern "C" void kernel_launch(void* const* d_in, const int* in_sizes, int n_in,
                              void* d_out, int out_size, void* d_ws, size_t ws_size,
                              hipStream_t stream)
{
    const float* x     = (const float*)d_in[0];
    const int*   ei    = (const int*)d_in[1];
    const int*   batch = (const int*)d_in[2];
    const float* Wl1 = (const float*)d_in[3];  const float* bl1 = (const float*)d_in[4];
    const float* Wr1 = (const float*)d_in[5];  const float* br1 = (const float*)d_in[6];
    const float* att1= (const float*)d_in[7];  const float* b1  = (const float*)d_in[8];
    const float* Wl2 = (const float*)d_in[9];  const float* bl2 = (const float*)d_in[10];
    const float* Wr2 = (const float*)d_in[11]; const float* br2 = (const float*)d_in[12];
    const float* att2= (const float*)d_in[13]; const float* b2  = (const float*)d_in[14];
    const float* Wrole=(const float*)d_in[15]; const float* brole=(const float*)d_in[16];
    const float* Wproj=(const float*)d_in[17]; const float* bproj=(const float*)d_in[18];

    const int n  = in_sizes[0] / 3;     // 20000
    const int E  = in_sizes[1] / 2;     // 320000
    const int EP = E + n;               // with self-loops
    const int G  = 64;
    const int* srcA = ei;
    const int* dstA = ei + E;

    // ---- workspace carve-up (float units), regions recycled across phases ----
    float* ws = (float*)d_ws;
    size_t off = 0;
    float* bufA = ws + off; off += (size_t)n * 256;   // xl1  -> later xl2|xr2
    float* bufB = ws + off; off += (size_t)n * 256;   // xr1  -> later agg2 + h2
    float* bufC = ws + off; off += (size_t)n * 256;   // agg1 -> h1 (in place)
    float* bufE = ws + off; off += (size_t)EP * 4;    // e1   -> later e2
    unsigned int* bufM = (unsigned int*)(ws + off); off += (size_t)n * 4;  // emax keys
    float* bufD = ws + off; off += (size_t)n * 4;     // denom
    float* gsum = ws + off; off += (size_t)G * 64;
    float* gcnt = ws + off; off += (size_t)G;

    const int TPB = 256;
    const int tot1 = EP * 4;            // conv1 edge-head items
    const int tot2 = EP * 2;            // conv2 edge-head items

    // ---------------- conv1 ----------------
    hipMemsetAsync(bufC, 0, (size_t)n * 256 * sizeof(float), stream);   // agg1 = 0
    hipMemsetAsync(bufM, 0, (size_t)n * 4 * sizeof(unsigned int), stream); // emax key 0 = -inf
    hipMemsetAsync(bufD, 0, (size_t)n * 4 * sizeof(float), stream);     // denom = 0
    hipMemsetAsync(gsum, 0, (size_t)(G * 64 + G) * sizeof(float), stream); // gsum+gcnt

    k_transform1<<<(n * 256 + TPB - 1) / TPB, TPB, 0, stream>>>(
        x, Wl1, bl1, Wr1, br1, bufA, bufB, n);
    k_edge_attn_max<4><<<(tot1 + TPB - 1) / TPB, TPB, 0, stream>>>(
        srcA, dstA, E, n, bufA, bufB, att1, bufE, bufM);
    k_edge_exp_sum<4><<<(tot1 + TPB - 1) / TPB, TPB, 0, stream>>>(
        dstA, E, n, bufE, bufM, bufD);
    k_edge_aggregate<4><<<(tot1 + TPB - 1) / TPB, TPB, 0, stream>>>(
        srcA, dstA, E, n, bufA, bufE, bufD, bufC);
    k_bias_relu<<<(n * 256 + TPB - 1) / TPB, TPB, 0, stream>>>(bufC, b1, n * 256, 256);
    // bufC now holds h1 [n,256]

    // ---------------- conv2 ----------------
    float* xl2  = bufA;
    float* xr2  = bufA + (size_t)n * 128;
    float* agg2 = bufB;
    float* h2   = bufB + (size_t)n * 128;
    hipMemsetAsync(agg2, 0, (size_t)n * 128 * sizeof(float), stream);
    hipMemsetAsync(bufM, 0, (size_t)n * 2 * sizeof(unsigned int), stream);
    hipMemsetAsync(bufD, 0, (size_t)n * 2 * sizeof(float), stream);

    const int mtiles = (n + 15) / 16;
    const int tiles_total = mtiles * 8 * 2;         // M tiles x 8 N-tiles x 2 weights
    k_transform2_wmma<<<(tiles_total + 3) / 4, 128, 0, stream>>>(
        bufC, Wl2, bl2, Wr2, br2, xl2, xr2, n, mtiles);
    k_edge_attn_max<2><<<(tot2 + TPB - 1) / TPB, TPB, 0, stream>>>(
        srcA, dstA, E, n, xl2, xr2, att2, bufE, bufM);
    k_edge_exp_sum<2><<<(tot2 + TPB - 1) / TPB, TPB, 0, stream>>>(
        dstA, E, n, bufE, bufM, bufD);
    k_edge_aggregate<2><<<(tot2 + TPB - 1) / TPB, TPB, 0, stream>>>(
        srcA, dstA, E, n, xl2, bufE, bufD, agg2);
    k_finalize2<<<(n * 64 + TPB - 1) / TPB, TPB, 0, stream>>>(agg2, b2, h2, n);

    // ---------------- heads ----------------
    float* out_roles = (float*)d_out;
    float* out_emb   = (float*)d_out + (size_t)n * 4;
    k_roles<<<(n * 4 + TPB - 1) / TPB, TPB, 0, stream>>>(h2, Wrole, brole, out_roles, n);
    k_pool<<<(n * 64 + TPB - 1) / TPB, TPB, 0, stream>>>(h2, batch, gsum, gcnt, n);
    k_proj<<<(G * 64 + TPB - 1) / TPB, TPB, 0, stream>>>(gsum, gcnt, Wproj, bproj, out_emb, G);
}